// QuantumPositionalEncoder_65481071407976
// MI455X (gfx1250) — compile-verified
//
#include <hip/hip_runtime.h>
#include <cstdint>

// Problem constants from the reference.
#define B_   16
#define S_   4096
#define D_   512
#define PER_LANE 16   // D_ / 32 lanes (wave32)

// Wave32 inclusive multiplicative scan (Hillis-Steele) via ds_bpermute.
__device__ __forceinline__ float wave_mul_scan_incl(float t, int lane) {
#pragma unroll
  for (int off = 1; off < 32; off <<= 1) {
    int src = lane - off;
    int idx = (src < 0 ? lane : src) << 2;          // byte index for bpermute
    int ui  = __builtin_amdgcn_ds_bpermute(idx, __float_as_int(t));
    float u = __int_as_float(ui);
    if (lane >= off) t *= u;
  }
  return t;
}

extern "C" __global__ void __launch_bounds__(256)
qpe_kernel(const float* __restrict__ x,
           const float* __restrict__ pe,
           const float* __restrict__ qwp,
           float* __restrict__ out) {
  __shared__ float s_pe[D_];

  const int s    = blockIdx.x;        // sequence position; pe row shared by all 16 batches
  const int tid  = threadIdx.x;
  const int lane = tid & 31;
  const int wave = tid >> 5;          // 8 waves per block

  // ---- gfx1250 async global->LDS copy of the 2KB pe row (256 threads x 8B) ----
  {
    const float* g   = pe + (size_t)s * D_ + tid * 2;
    uint32_t lds_addr = (uint32_t)(uintptr_t)(&s_pe[tid * 2]);
    uint64_t gaddr    = (uint64_t)(uintptr_t)g;
    asm volatile("global_load_async_to_lds_b64 %0, %1, off"
                 :: "v"(lds_addr), "v"(gaddr)
                 : "memory");
  }
#if __has_builtin(__builtin_amdgcn_s_wait_asynccnt)
  __builtin_amdgcn_s_wait_asynccnt(0);
#else
  asm volatile("s_wait_asynccnt 0" ::: "memory");
#endif
  __syncthreads();

  // pe slice for this lane (16 contiguous floats), reused for both rows.
  float per[PER_LANE];
  {
    const float4* p4 = (const float4*)&s_pe[lane * PER_LANE];
#pragma unroll
    for (int k = 0; k < 4; ++k) {
      float4 t = p4[k];
      per[4*k+0] = t.x; per[4*k+1] = t.y; per[4*k+2] = t.z; per[4*k+3] = t.w;
    }
  }

  // Scalar sigmoid blend weight (uniform -> scalar load).
  const float qw = qwp[0];
  const float w  = 1.0f / (1.0f + __expf(-qw));
  const float cw = 1.0f - w;

  // Each wave owns 2 rows: b = 2*wave + r. Issue both rows' loads up front.
  float v[2][PER_LANE];
#pragma unroll
  for (int r = 0; r < 2; ++r) {
    const int b = wave * 2 + r;
    const float4* g4 =
        (const float4*)(x + ((size_t)b * S_ + s) * D_ + lane * PER_LANE);
#pragma unroll
    for (int k = 0; k < 4; ++k) {
      float4 t = g4[k];
      v[r][4*k+0] = t.x; v[r][4*k+1] = t.y; v[r][4*k+2] = t.z; v[r][4*k+3] = t.w;
    }
  }

#pragma unroll
  for (int r = 0; r < 2; ++r) {
    const int b = wave * 2 + r;

    // Local (per-lane) serial cumprod over 16 contiguous elements.
    float o[PER_LANE], q[PER_LANE];
    float p = 1.0f;
#pragma unroll
    for (int j = 0; j < PER_LANE; ++j) {
      o[j] = v[r][j] + per[j];
      p *= __cosf(o[j]);
      q[j] = p;
    }

    // Cross-lane multiplicative scan; exclusive prefix = inclusive(lane-1).
    float incl = wave_mul_scan_incl(p, lane);
    int   pi   = __builtin_amdgcn_ds_bpermute(((lane > 0 ? lane - 1 : 0) << 2),
                                              __float_as_int(incl));
    float pref = (lane == 0) ? 1.0f : __int_as_float(pi);

    float4* o4 = (float4*)(out + ((size_t)b * S_ + s) * D_ + lane * PER_LANE);
#pragma unroll
    for (int k = 0; k < 4; ++k) {
      float4 t;
      t.x = w * (q[4*k+0] * pref) + cw * o[4*k+0];
      t.y = w * (q[4*k+1] * pref) + cw * o[4*k+1];
      t.z = w * (q[4*k+2] * pref) + cw * o[4*k+2];
      t.w = w * (q[4*k+3] * pref) + cw * o[4*k+3];
      o4[k] = t;
    }
  }
}

extern "C" void kernel_launch(void* const* d_in, const int* in_sizes, int n_in,
                              void* d_out, int out_size, void* d_ws, size_t ws_size,
                              hipStream_t stream) {
  (void)in_sizes; (void)n_in; (void)out_size; (void)d_ws; (void)ws_size;
  const float* x   = (const float*)d_in[0];   // [16, 4096, 512] f32
  const float* pe  = (const float*)d_in[1];   // [1, 5000, 512] f32
  // d_in[2] = param (provably unused by the reference)
  const float* qw  = (const float*)d_in[3];   // [1] f32
  float* out = (float*)d_out;                 // [16, 4096, 512] f32

  dim3 grid(S_);     // one block per sequence position (pe row shared 16x)
  dim3 block(256);   // 8 waves; each wave scans 2 rows of 512
  qpe_kernel<<<grid, block, 0, stream>>>(x, pe, qw, out);
}